// Pipeline_8400956031319
// MI455X (gfx1250) — compile-verified
//
#include <hip/hip_runtime.h>
#include <stdint.h>

// NMS pipeline for MI455X (gfx1250, wave32).
// Phase A: pack boxes -> aligned float4[8192]
// Phase B: pairwise IoU suppression bitmask, double-buffered
//          global_load_async_to_lds_b128 staging (ASYNCcnt pipelining)
// Phase C: blocked greedy scan: 32 rows per step, diagonal-block serial
//          resolution in registers + uniform-branch row ORs (no per-row barrier)
// Phase D: argmax + masked writes of the three outputs.

#define N_BOX 8192
#define WORDS (N_BOX / 32)       // 256 mask words per row
#define IOU_T 0.6f

__device__ __forceinline__ unsigned lds_byte_off(const void* p) {
    // Generic pointer to __shared__ is {SHARED_BASE, lds_offset}; low 32 bits
    // are the LDS byte address (CDNA5 aperture rule).
    return (unsigned)(uintptr_t)p;
}

// ---------------- Phase A: gather boxes into packed, 16B-aligned float4 ----
__global__ void pack_boxes(const float* __restrict__ src, int stride,
                           float4* __restrict__ dst) {
    int i = blockIdx.x * blockDim.x + threadIdx.x;
    if (i < N_BOX) {
        const float* r = src + (size_t)i * stride + 1;   // cols 1..4 = x1,y1,x2,y2
        dst[i] = make_float4(r[0], r[1], r[2], r[3]);
    }
}

// ---------------- Phase B: IoU suppression mask -----------------------------
// Block = 32 rows, blockDim = 256 (8 wave32s). Column boxes staged through two
// LDS buffers with async-to-LDS DMA; s_wait_asynccnt 1 keeps one chunk in flight.
__global__ __launch_bounds__(256) void iou_mask(const float4* __restrict__ boxes,
                                                unsigned* __restrict__ mask) {
    __shared__ float4 rowBox[32];
    __shared__ float  rowArea[32];
    __shared__ float4 colBuf[2][256];

    const int t = threadIdx.x;
    const int rowBase = blockIdx.x * 32;

    if (t < 32) {
        float4 b = boxes[rowBase + t];
        rowBox[t]  = b;
        rowArea[t] = fmaxf(b.z - b.x, 0.0f) * fmaxf(b.w - b.y, 0.0f);
    }

    // Prologue: async-stage chunk 0 into buffer 0 (each lane pulls one float4).
    {
        unsigned ldsDst = lds_byte_off(&colBuf[0][t]);
        unsigned vOff   = (unsigned)(t * 16);
        asm volatile("global_load_async_to_lds_b128 %0, %1, %2"
                     :: "v"(ldsDst), "v"(vOff), "s"(boxes) : "memory");
    }

    const int r    = t >> 3;        // row within block: 0..31
    const int wsub = t & 7;         // word within 256-column chunk: 0..7
    const int i    = rowBase + r;

    __syncthreads();                // rowBox/rowArea visible
    const float4 a     = rowBox[r];
    const float  areaA = rowArea[r];

    const int NCHUNK = N_BOX / 256; // 32
    for (int c = 0; c < NCHUNK; ++c) {
        const int buf = c & 1;
        // Issue next chunk into the alternate buffer (its last readers finished
        // at iteration c-1, protected by the end-of-iteration barrier).
        if (c + 1 < NCHUNK) {
            unsigned ldsDst = lds_byte_off(&colBuf[buf ^ 1][t]);
            unsigned vOff   = (unsigned)(((c + 1) * 256 + t) * 16);
            asm volatile("global_load_async_to_lds_b128 %0, %1, %2"
                         :: "v"(ldsDst), "v"(vOff), "s"(boxes) : "memory");
            asm volatile("s_wait_asynccnt 1" ::: "memory"); // chunk c landed (in-order)
        } else {
            asm volatile("s_wait_asynccnt 0" ::: "memory");
        }
        __syncthreads();            // cross-wave visibility of chunk c

        const int chunk = c * 256;
        unsigned bits = 0u;
#pragma unroll 8
        for (int k = 0; k < 32; ++k) {
            int    j = chunk + wsub * 32 + k;
            float4 b = colBuf[buf][wsub * 32 + k];
            float areaB = fmaxf(b.z - b.x, 0.0f) * fmaxf(b.w - b.y, 0.0f);
            float ix1 = fmaxf(a.x, b.x);
            float iy1 = fmaxf(a.y, b.y);
            float ix2 = fminf(a.z, b.z);
            float iy2 = fminf(a.w, b.w);
            float inter = fmaxf(ix2 - ix1, 0.0f) * fmaxf(iy2 - iy1, 0.0f);
            float uni   = areaA + areaB - inter;
            float iou   = inter / fmaxf(uni, 1e-9f);
            if ((j > i) && (iou > IOU_T)) bits |= (1u << k);
        }
        mask[(size_t)i * WORDS + (chunk >> 5) + wsub] = bits;
        __syncthreads();            // buffer-reuse guard for next iteration's issue
    }
}

// ---------------- Phase C: blocked greedy scan ------------------------------
// One workgroup; thread t owns suppressed word t (columns t*32..t*32+31).
// Per 32-row block: resolve in-block greedy order serially on the 32x32
// diagonal words (single lane, register ops), broadcast kept mask, then OR the
// kept rows' words t>b with uniform (scalar) branches — no per-row barriers.
__global__ __launch_bounds__(256) void nms_scan(const unsigned* __restrict__ mask,
                                                unsigned* __restrict__ keepOut) {
    __shared__ unsigned rem[WORDS];   // suppressed bitset
    __shared__ unsigned diag[32];     // diagonal-block words of current row block
    __shared__ unsigned keptShared;

    const int t = threadIdx.x;
    rem[t] = 0u;
    __syncthreads();

    for (int b = 0; b < WORDS; ++b) {             // rows [32b, 32b+32)
        if (t < 32)
            diag[t] = mask[(size_t)(32 * b + t) * WORDS + b];
        __syncthreads();

        if (t == 0) {
            unsigned s = rem[b];                  // suppression from earlier rows
#pragma unroll
            for (int k = 0; k < 32; ++k)
                if (!((s >> k) & 1u)) s |= diag[k];   // greedy in-block order
            keptShared = ~s;                      // final keep bits for this block
            rem[b] = s;                           // word b fully resolved
        }
        __syncthreads();
        const unsigned kept = keptShared;         // wave-uniform

        if (t > b) {                              // rows only set bits j > i
            unsigned acc = rem[t];
#pragma unroll 8
            for (int k = 0; k < 32; ++k) {
                if ((kept >> k) & 1u)             // uniform -> scalar branch
                    acc |= mask[(size_t)(32 * b + k) * WORDS + t];
            }
            rem[t] = acc;
        }
        __syncthreads();
    }
    keepOut[t] = ~rem[t];
}

// ---------------- Phase D: argmax + masked outputs --------------------------
__global__ void finalize(const float* __restrict__ det, const float* __restrict__ rpn,
                         const unsigned* __restrict__ keepDet,
                         const unsigned* __restrict__ keepRpn,
                         float* __restrict__ validOut, float* __restrict__ invalidOut,
                         float* __restrict__ rpnOut) {
    int i = blockIdx.x * blockDim.x + threadIdx.x;
    if (i >= N_BOX) return;

    bool kd = (keepDet[i >> 5] >> (i & 31)) & 1u;
    const float* d = det + (size_t)i * 9;
    float s0 = d[5], s1 = d[6], s2 = d[7], s3 = d[8];
    int am = 0; float best = s0;                 // first-max semantics = jnp.argmax
    if (s1 > best) { best = s1; am = 1; }
    if (s2 > best) { best = s2; am = 2; }
    if (s3 > best) { best = s3; am = 3; }
    float vm = (kd && am != 0) ? 1.0f : 0.0f;
    float im = (kd && am == 0) ? 1.0f : 0.0f;
#pragma unroll
    for (int c = 0; c < 9; ++c) {
        validOut[(size_t)i * 9 + c]   = d[c] * vm;
        invalidOut[(size_t)i * 9 + c] = d[c] * im;
    }

    bool kr = (keepRpn[i >> 5] >> (i & 31)) & 1u;
    float rm = kr ? 1.0f : 0.0f;
    const float* rp = rpn + (size_t)i * 6;
#pragma unroll
    for (int c = 0; c < 6; ++c)
        rpnOut[(size_t)i * 6 + c] = rp[c] * rm;
}

extern "C" void kernel_launch(void* const* d_in, const int* in_sizes, int n_in,
                              void* d_out, int out_size, void* d_ws, size_t ws_size,
                              hipStream_t stream) {
    (void)in_sizes; (void)n_in; (void)out_size; (void)ws_size;

    const float* det = (const float*)d_in[0];   // (8192, 9)
    const float* rpn = (const float*)d_in[1];   // (8192, 6)

    // Workspace layout (stream-ordered reuse keeps it ~8.3 MB):
    //   [0, 8MB)            suppression mask (reused det -> rpn)
    //   [8MB, 8MB+128KB)    packed float4 boxes (reused det -> rpn)
    //   then keepDet[256], keepRpn[256] u32 words
    char* ws = (char*)d_ws;
    const size_t maskBytes  = (size_t)N_BOX * WORDS * sizeof(unsigned);  // 8 MB
    const size_t boxBytes   = (size_t)N_BOX * sizeof(float4);            // 128 KB
    unsigned* mask    = (unsigned*)ws;
    float4*   boxes   = (float4*)(ws + maskBytes);
    unsigned* keepDet = (unsigned*)(ws + maskBytes + boxBytes);
    unsigned* keepRpn = keepDet + WORDS;

    float* validOut   = (float*)d_out;
    float* invalidOut = validOut + (size_t)N_BOX * 9;
    float* rpnOut     = invalidOut + (size_t)N_BOX * 9;

    // Detections NMS
    pack_boxes<<<N_BOX / 256, 256, 0, stream>>>(det, 9, boxes);
    iou_mask  <<<N_BOX / 32,  256, 0, stream>>>(boxes, mask);
    nms_scan  <<<1,           256, 0, stream>>>(mask, keepDet);

    // RPN NMS (reuses mask/box scratch; stream ordering serializes)
    pack_boxes<<<N_BOX / 256, 256, 0, stream>>>(rpn, 6, boxes);
    iou_mask  <<<N_BOX / 32,  256, 0, stream>>>(boxes, mask);
    nms_scan  <<<1,           256, 0, stream>>>(mask, keepRpn);

    finalize<<<N_BOX / 256, 256, 0, stream>>>(det, rpn, keepDet, keepRpn,
                                              validOut, invalidOut, rpnOut);
}